// DiffusionNet3D_62637803045177
// MI455X (gfx1250) — compile-verified
//
#include <hip/hip_runtime.h>
#include <hip/hip_bf16.h>

// Problem constants (from reference)
constexpr int NB   = 32;            // batch
constexpr int ND   = 48;            // depth
constexpr int NH   = 48;            // height
constexpr int NW   = 48;            // width
constexpr int NPOS = ND * NH * NW;  // 110592 spatial positions
constexpr int NCLS = 10;
constexpr int TSTEPS = 64;          // reference T (python scalar, fixed)

typedef __attribute__((ext_vector_type(2))) float v2f;
typedef __attribute__((ext_vector_type(8))) float v8f;

// gfx1250 hardware tanh (transcendental VALU op). Builtin preferred so the
// backend handles TRANS-op hazards; asm fallback includes an independent op
// after the trans instruction per the CDNA5 hazard rule.
__device__ __forceinline__ float hw_tanh(float x) {
#if __has_builtin(__builtin_amdgcn_tanhf)
  return __builtin_amdgcn_tanhf(x);
#else
  float y;
  asm volatile("v_tanh_f32 %0, %1\n\tv_nop" : "=v"(y) : "v"(x));
  return y;
#endif
}

// ---------------------------------------------------------------------------
// Init: state0 in batch-innermost layout S[pos][b]; plane d==0 <- x[b,0,h,w]
// ---------------------------------------------------------------------------
__global__ void dn3d_init(const float* __restrict__ x, float* __restrict__ s) {
  size_t i = (size_t)blockIdx.x * blockDim.x + threadIdx.x;
  if (i >= (size_t)NPOS * NB) return;
  int b   = (int)(i & (NB - 1));
  int pos = (int)(i >> 5);
  int d   = pos / (NH * NW);
  float v = 0.0f;
  if (d == 0) {
    int hw = pos % (NH * NW);
    v = x[(size_t)b * (NH * NW) + hw];
  }
  s[i] = v;
}

// ---------------------------------------------------------------------------
// One diffusion step. Layout S[pos][b], pos = (d*48+h)*48+w, b innermost so a
// wave32 covers all 32 batches at one position: neighbor loads are single
// 128B cachelines, filter values are wave-uniform (SMEM s_loads).
// Each wave handles a half-row (24 consecutive w) with a rolling 3-column
// register window -> 9 vector loads + 27 FMAs + 1 tanh per output column.
// ---------------------------------------------------------------------------
__global__ __launch_bounds__(256) void dn3d_step(const float* __restrict__ in,
                                                 float* __restrict__ out,
                                                 const float* __restrict__ filt) {
  const int lane = threadIdx.x & 31;
  const int wid  = threadIdx.x >> 5;
  // wave-uniform work id (forced into SGPR so filter loads become scalar)
  const int gw   = __builtin_amdgcn_readfirstlane((int)blockIdx.x * 8 + wid);
  const int half = gw & 1;        // which half of the W row
  const int dh   = gw >> 1;       // 0..2303 = d*48+h
  const int hh   = dh % NH;
  const int dd   = dh / NH;
  const int w0   = half * (NW / 2);

  // 9 (dz,dy) row bases + validity (zero padding outside the volume)
  int  rbase[9];
  bool rval[9];
#pragma unroll
  for (int dz = 0; dz < 3; ++dz) {
#pragma unroll
    for (int dy = 0; dy < 3; ++dy) {
      int zz = dd + dz - 1, yy = hh + dy - 1;
      bool v = (zz >= 0) && (zz < ND) && (yy >= 0) && (yy < NH);
      rval[dz * 3 + dy]  = v;
      rbase[dz * 3 + dy] = ((zz * NH + yy) * NW) * NB;
    }
  }

  float p[9], c[9], n[9];

  auto loadcol = [&](int w, float col[9]) {
    const bool wv = (w >= 0) && (w < NW);
#pragma unroll
    for (int j = 0; j < 9; ++j)
      col[j] = (wv && rval[j]) ? in[(size_t)rbase[j] + (size_t)w * NB + lane] : 0.0f;
  };

  loadcol(w0 - 1, p);
  loadcol(w0,     c);

  // wave-uniform filter pointer: 27 consecutive fp32 per position
  const float* fp = filt + (size_t)((dd * NH + hh) * NW + w0) * 27;

  for (int i = 0; i < NW / 2; ++i) {
    const int w = w0 + i;
    loadcol(w + 1, n);
    float acc = 0.0f;
#pragma unroll
    for (int j = 0; j < 9; ++j) {
      acc = fmaf(fp[j * 3 + 0], p[j], acc);   // dx = 0  (w-1)
      acc = fmaf(fp[j * 3 + 1], c[j], acc);   // dx = 1  (w)
      acc = fmaf(fp[j * 3 + 2], n[j], acc);   // dx = 2  (w+1)
    }
    out[(size_t)((dd * NH + hh) * NW + w) * NB + lane] = hw_tanh(acc);
    fp += 27;
#pragma unroll
    for (int j = 0; j < 9; ++j) { p[j] = c[j]; c[j] = n[j]; }
  }
}

// ---------------------------------------------------------------------------
// Pooling as a GEMM on the tensor core:
//   logits[b, cls] = sum_{k=h*48+w} plane[b, k] * Wt[k, cls]
// with Wt[k, cls] = 1/(48*binwidth) if h in bin(cls), else 0 (batch-shared).
// M = 32 batches (two 16-tiles, one per wave), N = 16 (10 classes + pad),
// K = 2304 -> 576 x V_WMMA_F32_16X16X4_F32 per wave, full fp32 precision.
// ---------------------------------------------------------------------------
__global__ __launch_bounds__(64) void dn3d_pool(const float* __restrict__ s,
                                                float* __restrict__ out) {
  const int lane = threadIdx.x & 31;
  const int tile = threadIdx.x >> 5;   // 0: batches 0-15, 1: batches 16-31
  const int half = lane >> 4;          // lanes 0-15 vs 16-31
  const int l15  = lane & 15;
  const int bb   = tile * 16 + l15;    // A-matrix row (batch)
  const int cls  = l15;                // B-matrix column (class)

  // adaptive bins over H=48 (computed branchlessly; masked for cls >= 10)
  const int sc = (cls * NH) / NCLS;
  const int ec = ((cls + 1) * NH + NCLS - 1) / NCLS;
  const float invw = 1.0f / ((float)NW * (float)(ec - sc));

  const int planePos = (ND - 1) * NH * NW;   // pos of (d=47, h=0, w=0)

  v8f acc = {};
  for (int k0 = 0; k0 < NH * NW; k0 += 4) {
    const int ka = k0 + half * 2;            // this half-wave's K pair
    // A: 16x4 f32, lane layout per ISA (lanes 0-15: K=0,1; lanes 16-31: K=2,3)
    v2f A;
    A.x = s[(size_t)(planePos + ka)     * NB + bb];
    A.y = s[(size_t)(planePos + ka + 1) * NB + bb];
    // B: 4x16 f32 weight matrix built on the fly (wave-uniform K, per-lane N)
    const int h0 = ka / NW, h1 = (ka + 1) / NW;
    v2f B;
    B.x = (cls < NCLS && h0 >= sc && h0 < ec) ? invw : 0.0f;
    B.y = (cls < NCLS && h1 >= sc && h1 < ec) ? invw : 0.0f;
    acc = __builtin_amdgcn_wmma_f32_16x16x4_f32(
        /*neg_a=*/false, A, /*neg_b=*/false, B,
        /*c_mod=*/(short)0, acc, /*reuse_a=*/false, /*reuse_b=*/false);
  }

  // D layout: VGPR r -> M = r (lanes 0-15) / M = r+8 (lanes 16-31), N = lane&15
#pragma unroll
  for (int r = 0; r < 8; ++r) {
    const int row = tile * 16 + half * 8 + r;
    if (cls < NCLS) out[row * NCLS + cls] = acc[r];
  }
}

// ---------------------------------------------------------------------------
// Host launcher: init -> 64 ping-pong steps -> WMMA pooling. Needs ~28.4 MB
// of d_ws for the two state buffers (everything stays L2-resident on MI455X).
// ---------------------------------------------------------------------------
extern "C" void kernel_launch(void* const* d_in, const int* in_sizes, int n_in,
                              void* d_out, int out_size, void* d_ws, size_t ws_size,
                              hipStream_t stream) {
  const float* x    = (const float*)d_in[0];   // (32,1,48,48) fp32
  const float* filt = (const float*)d_in[1];   // (48,48,48,3,3,3) fp32
  (void)in_sizes; (void)n_in; (void)out_size; (void)ws_size;

  float* s0 = (float*)d_ws;
  float* s1 = s0 + (size_t)NPOS * NB;

  {
    const size_t total = (size_t)NPOS * NB;
    dn3d_init<<<(unsigned)((total + 255) / 256), 256, 0, stream>>>(x, s0);
  }

  float* cur = s0;
  float* nxt = s1;
  const int nwaves = 2 * ND * NH;          // 4608 waves, half-row each
  const int blocks = nwaves / 8;           // 256-thread blocks = 8 waves
  for (int t = 0; t < TSTEPS; ++t) {
    dn3d_step<<<blocks, 256, 0, stream>>>(cur, nxt, filt);
    float* tmp = cur; cur = nxt; nxt = tmp;
  }
  // TSTEPS is even -> final state is back in s0 (== cur)

  dn3d_pool<<<1, 64, 0, stream>>>(cur, (float*)d_out);
}